// HistLoss_71159018160893
// MI455X (gfx1250) — compile-verified
//
#include <hip/hip_runtime.h>
#include <stdint.h>

#define CH 64
#define HW (512 * 512)
#define STRENGTH_F 100.0f

typedef __attribute__((ext_vector_type(2))) float v2f;
typedef __attribute__((ext_vector_type(8))) float v8f;

// ---------------- workspace layout (byte offsets) ----------------
// 0      : stats[2]        u32 {nI, nJ}
// 256    : minBits[64]     u32 (float bits; ordered compare valid, values >= 0)
// 512    : maxBits[64]     u32
// 1024   : hisRaw[64*256]  u32 (unscaled masked histogram of target)
// 66560  : cumJ[64*256]    f32 (cumsum scaled by nI/nJ)
// 132096 : partials[4096]  f32
// 148480 : rankBase[64*nbins] u32
#define OFF_STATS 0
#define OFF_MIN   256
#define OFF_MAX   512
#define OFF_HIS   1024
#define OFF_CUM   66560
#define OFF_PART  132096
#define OFF_RANK  148480

// ---------------------------------------------------------------- utils
__global__ void zero_ws_kernel(uint32_t* p, unsigned n) {
  unsigned i = blockIdx.x * blockDim.x + threadIdx.x;
  unsigned stride = gridDim.x * blockDim.x;
  for (; i < n; i += stride) p[i] = 0u;
}

__global__ void init_minmax_kernel(uint32_t* minBits) {
  if (threadIdx.x < CH) minBits[threadIdx.x] = 0x7F800000u;  // +inf
}

// ---------------------------------------------------------------- nI, nJ
__global__ __launch_bounds__(256)
void mask_count_kernel(const int* mI, const int* mJ, uint32_t* stats) {
  __shared__ uint32_t sI[256], sJ[256];
  uint32_t aI = 0, aJ = 0;
  for (int i = blockIdx.x * 256 + threadIdx.x; i < HW; i += gridDim.x * 256) {
    aI += (mI[i] > 0) ? 1u : 0u;
    aJ += (mJ[i] > 0) ? 1u : 0u;
  }
  sI[threadIdx.x] = aI; sJ[threadIdx.x] = aJ; __syncthreads();
  for (int off = 128; off > 0; off >>= 1) {
    if ((int)threadIdx.x < off) {
      sI[threadIdx.x] += sI[threadIdx.x + off];
      sJ[threadIdx.x] += sJ[threadIdx.x + off];
    }
    __syncthreads();
  }
  if (threadIdx.x == 0) { atomicAdd(&stats[0], sI[0]); atomicAdd(&stats[1], sJ[0]); }
}

// ---------------------------------------------------------------- masked min/max of target
__global__ __launch_bounds__(256)
void minmax_kernel(const float* J, const int* mJ, uint32_t* minBits, uint32_t* maxBits) {
  const int c = blockIdx.y;
  const float* Jc = J + (size_t)c * HW;
  const int chunk = HW / gridDim.x;
  const int start = blockIdx.x * chunk;
  uint32_t mn = 0x7F800000u, mx = 0u;
  for (int i = start + threadIdx.x; i < start + chunk; i += 256) {
    if (mJ[i] > 0) {
      uint32_t b = __float_as_uint(Jc[i]);  // nonnegative floats: bit order == value order
      mn = mn < b ? mn : b;
      mx = mx > b ? mx : b;
    }
  }
  __shared__ uint32_t smn[256], smx[256];
  smn[threadIdx.x] = mn; smx[threadIdx.x] = mx; __syncthreads();
  for (int off = 128; off > 0; off >>= 1) {
    if ((int)threadIdx.x < off) {
      smn[threadIdx.x] = smn[threadIdx.x] < smn[threadIdx.x + off] ? smn[threadIdx.x] : smn[threadIdx.x + off];
      smx[threadIdx.x] = smx[threadIdx.x] > smx[threadIdx.x + off] ? smx[threadIdx.x] : smx[threadIdx.x + off];
    }
    __syncthreads();
  }
  if (threadIdx.x == 0) { atomicMin(&minBits[c], smn[0]); atomicMax(&maxBits[c], smx[0]); }
}

// ---------------------------------------------------------------- masked 256-bin histogram of target
__global__ __launch_bounds__(256)
void histJ_kernel(const float* J, const int* mJ, const uint32_t* minBits,
                  const uint32_t* maxBits, uint32_t* hisRaw) {
  const int c = blockIdx.y;
  __shared__ uint32_t h[256];
  h[threadIdx.x] = 0u; __syncthreads();
  const float mn = __uint_as_float(minBits[c]);
  const float mx = __uint_as_float(maxBits[c]);
  const float step = (mx - mn) * (1.0f / 256.0f);
  const float inv = 1.0f / fmaxf(step, 1e-12f);
  const float* Jc = J + (size_t)c * HW;
  const int chunk = HW / gridDim.x;
  const int start = blockIdx.x * chunk;
  for (int i = start + threadIdx.x; i < start + chunk; i += 256) {
    if (mJ[i] > 0) {
      int b = (int)floorf((Jc[i] - mn) * inv);
      b = min(max(b, 0), 255);
      atomicAdd(&h[b], 1u);
    }
  }
  __syncthreads();
  atomicAdd(&hisRaw[c * 256 + threadIdx.x], h[threadIdx.x]);
}

// ---------------------------------------------------------------- cumsum via WMMA
// cum(256 x 16ch) = L256(lower-tri ones) * H(256 x 16ch), tiled 16x16:
//   cum_tile_I = Lincl * H_I + Ones * (sum_{K<I} H_K)
// f32 16x16x4 WMMA, K chained in chunks of 4. Counts are ints < 2^24 -> exact.
__global__ __launch_bounds__(32)
void cumsum_wmma_kernel(const uint32_t* hisRaw, const uint32_t* stats, float* cumJ) {
  const int lane = threadIdx.x;
  const int n = lane & 15;          // column = channel-in-tile, also A-row m
  const bool hilane = lane >= 16;
  const int ch = blockIdx.x * 16 + n;
  const float nIf = (float)stats[0];
  const float nJf = (float)stats[1];
  const float scale = nIf / fmaxf(nJf, 1.0f);

  v2f S[4];                         // running column sums, B layout
  #pragma unroll
  for (int q = 0; q < 4; ++q) { S[q].x = 0.0f; S[q].y = 0.0f; }
  v2f ones; ones.x = 1.0f; ones.y = 1.0f;

  for (int I = 0; I < 16; ++I) {
    v2f Hc[4];
    #pragma unroll
    for (int q = 0; q < 4; ++q) {   // B layout: v0 holds K=4q+(hi?2:0), v1 K=4q+(hi?3:1)
      const int k0 = 4 * q + (hilane ? 2 : 0);
      const int k1 = 4 * q + (hilane ? 3 : 1);
      Hc[q].x = (float)hisRaw[ch * 256 + 16 * I + k0];
      Hc[q].y = (float)hisRaw[ch * 256 + 16 * I + k1];
    }
    v8f acc = {0.f, 0.f, 0.f, 0.f, 0.f, 0.f, 0.f, 0.f};
    #pragma unroll
    for (int q = 0; q < 4; ++q) {   // Lincl: A[m][k] = (k <= m)
      const int k0 = 4 * q + (hilane ? 2 : 0);
      const int k1 = 4 * q + (hilane ? 3 : 1);
      v2f a; a.x = (k0 <= n) ? 1.0f : 0.0f; a.y = (k1 <= n) ? 1.0f : 0.0f;
      acc = __builtin_amdgcn_wmma_f32_16x16x4_f32(false, a, false, Hc[q],
                                                  (short)0, acc, false, false);
    }
    #pragma unroll
    for (int q = 0; q < 4; ++q) {   // Ones * running-sum
      acc = __builtin_amdgcn_wmma_f32_16x16x4_f32(false, ones, false, S[q],
                                                  (short)0, acc, false, false);
    }
    #pragma unroll
    for (int r = 0; r < 8; ++r) {   // D layout: VGPR r = row r (lanes 0-15) / r+8 (16-31)
      const int row = r + (hilane ? 8 : 0);
      cumJ[ch * 256 + 16 * I + row] = acc[r] * scale;
    }
    #pragma unroll
    for (int q = 0; q < 4; ++q) { S[q].x += Hc[q].x; S[q].y += Hc[q].y; }
  }
}

// ---------------------------------------------------------------- fine rank histogram of masked input
__global__ __launch_bounds__(256)
void rank_hist_kernel(const float* In, const int* mI, uint32_t* rankHist, int nbins) {
  const int c = blockIdx.y;
  const float* Ic = In + (size_t)c * HW;
  uint32_t* h = rankHist + (size_t)c * nbins;
  const float fb = (float)nbins;
  const int chunk = HW / gridDim.x;
  const int start = blockIdx.x * chunk;
  for (int i = start + threadIdx.x; i < start + chunk; i += 256) {
    if (mI[i] > 0) {
      int b = (int)floorf(Ic[i] * fb);
      b = min(max(b, 0), nbins - 1);
      atomicAdd(&h[b], 1u);
    }
  }
}

// per-channel exclusive scan -> first rank per bin (1-based)
__global__ __launch_bounds__(1024)
void rank_scan_kernel(uint32_t* rankHist, int nbins) {
  const int c = blockIdx.x;
  uint32_t* h = rankHist + (size_t)c * nbins;
  __shared__ uint32_t sh[1024];
  uint32_t running = 0;
  for (int base = 0; base < nbins; base += 1024) {
    const uint32_t orig = h[base + threadIdx.x];
    sh[threadIdx.x] = orig; __syncthreads();
    for (int off = 1; off < 1024; off <<= 1) {
      const uint32_t t = ((int)threadIdx.x >= off) ? sh[threadIdx.x - off] : 0u;
      __syncthreads();
      sh[threadIdx.x] += t;
      __syncthreads();
    }
    const uint32_t incl = sh[threadIdx.x];
    h[base + threadIdx.x] = running + (incl - orig) + 1u;  // exclusive + 1
    const uint32_t tot = sh[1023];
    __syncthreads();
    running += tot;
  }
}

// ---------------------------------------------------------------- rank -> CDF-inverse remap -> sq-err
__global__ __launch_bounds__(256)
void loss_kernel(const float* In, const int* mI, const float* cumJ,
                 const uint32_t* minBits, const uint32_t* maxBits,
                 uint32_t* rankBase, int nbins, float* partials) {
  const int c = blockIdx.y;
  __shared__ float cum[256];
  cum[threadIdx.x] = cumJ[c * 256 + threadIdx.x];
  __syncthreads();
  const float mn = __uint_as_float(minBits[c]);
  const float mx = __uint_as_float(maxBits[c]);
  const float step = (mx - mn) * (1.0f / 256.0f);
  const float* Ic = In + (size_t)c * HW;
  uint32_t* rb = rankBase + (size_t)c * nbins;
  const float fb = (float)nbins;
  const int chunk = HW / gridDim.x;
  const int start = blockIdx.x * chunk;
  float acc = 0.0f;
  for (int i = start + threadIdx.x; i < start + chunk; i += 256) {
    if (mI[i] > 0) {
      const float v = Ic[i];
      int kb = (int)floorf(v * fb);
      kb = min(max(kb, 0), nbins - 1);
      const float rf = (float)atomicAdd(&rb[kb], 1u);  // rank in 1..nI
      // searchsorted(cum, rf, 'left') over 256 bins
      int lo = 0, hi = 256;
      while (lo < hi) { const int mid = (lo + hi) >> 1; if (cum[mid] < rf) lo = mid + 1; else hi = mid; }
      const int b = min(lo, 255);
      const float cb   = cum[b];
      const float cbm1 = (b > 0) ? cum[b - 1] : 0.0f;
      const float hisb = cb - cbm1;                    // == his[b] exactly
      float ratio = (rf - cbm1) / fmaxf(hisb, 1e-12f);
      ratio = fminf(fmaxf(ratio, 0.0f), 1.0f);
      const float val = mn + ((float)b + ratio) * step;
      const float d = v - val;
      acc += d * d;
    }
  }
  __shared__ float s[256];
  s[threadIdx.x] = acc; __syncthreads();
  for (int off = 128; off > 0; off >>= 1) {
    if ((int)threadIdx.x < off) s[threadIdx.x] += s[threadIdx.x + off];
    __syncthreads();
  }
  if (threadIdx.x == 0) partials[blockIdx.y * gridDim.x + blockIdx.x] = s[0];
}

__global__ __launch_bounds__(256)
void final_kernel(const float* partials, int n, float* out) {
  __shared__ float s[256];
  float a = 0.0f;
  for (int i = threadIdx.x; i < n; i += 256) a += partials[i];
  s[threadIdx.x] = a; __syncthreads();
  for (int off = 128; off > 0; off >>= 1) {
    if ((int)threadIdx.x < off) s[threadIdx.x] += s[threadIdx.x + off];
    __syncthreads();
  }
  if (threadIdx.x == 0) out[0] = s[0] * (STRENGTH_F / (float)((size_t)CH * HW));
}

// ---------------------------------------------------------------- launch
extern "C" void kernel_launch(void* const* d_in, const int* in_sizes, int n_in,
                              void* d_out, int out_size, void* d_ws, size_t ws_size,
                              hipStream_t stream) {
  (void)in_sizes; (void)n_in; (void)out_size;
  const float* In = (const float*)d_in[0];
  const float* Tg = (const float*)d_in[1];
  const int*   mI = (const int*)d_in[2];
  const int*   mJ = (const int*)d_in[3];
  float* out = (float*)d_out;

  char* ws = (char*)d_ws;
  uint32_t* stats   = (uint32_t*)(ws + OFF_STATS);
  uint32_t* minBits = (uint32_t*)(ws + OFF_MIN);
  uint32_t* maxBits = (uint32_t*)(ws + OFF_MAX);
  uint32_t* hisRaw  = (uint32_t*)(ws + OFF_HIS);
  float*    cumJ    = (float*)   (ws + OFF_CUM);
  float*    parts   = (float*)   (ws + OFF_PART);
  uint32_t* rankB   = (uint32_t*)(ws + OFF_RANK);

  // fine-rank bin count, shrink to fit workspace (power of two, >= 1024)
  int nbins = 65536;
  while (nbins > 1024 && (size_t)OFF_RANK + (size_t)CH * nbins * 4 > ws_size) nbins >>= 1;

  const unsigned zero_words = (unsigned)((OFF_RANK + (size_t)CH * nbins * 4) / 4);
  zero_ws_kernel<<<2048, 256, 0, stream>>>((uint32_t*)ws, zero_words);
  init_minmax_kernel<<<1, 64, 0, stream>>>(minBits);

  mask_count_kernel<<<256, 256, 0, stream>>>(mI, mJ, stats);
  minmax_kernel<<<dim3(32, CH), 256, 0, stream>>>(Tg, mJ, minBits, maxBits);
  histJ_kernel<<<dim3(32, CH), 256, 0, stream>>>(Tg, mJ, minBits, maxBits, hisRaw);
  cumsum_wmma_kernel<<<CH / 16, 32, 0, stream>>>(hisRaw, stats, cumJ);
  rank_hist_kernel<<<dim3(32, CH), 256, 0, stream>>>(In, mI, rankB, nbins);
  rank_scan_kernel<<<CH, 1024, 0, stream>>>(rankB, nbins);
  loss_kernel<<<dim3(64, CH), 256, 0, stream>>>(In, mI, cumJ, minBits, maxBits,
                                                rankB, nbins, parts);
  final_kernel<<<1, 256, 0, stream>>>(parts, 64 * CH, out);
}